// LinearSplineKANLayer_66718021976512
// MI455X (gfx1250) — compile-verified
//
#include <hip/hip_runtime.h>
#include <hip/hip_bf16.h>
#include <math.h>

// ---- problem constants ------------------------------------------------------
#define IN_F    64
#define OUT_F   64
#define GRIDSZ  17
#define BATCH   8192
#define KSP     (IN_F * GRIDSZ)      // 1088 = spline K dimension
#define GRID_LO (-2.5f)
#define H       0.3125f              // 5/16, exact in fp32
#define INV_H   3.2f
#define INV_SQRT_IN 0.125f           // 1/sqrt(64)

typedef __attribute__((ext_vector_type(2))) float v2f;
typedef __attribute__((ext_vector_type(8))) float v8f;

// silu via raw hardware v_rcp_f32 (1 ulp) instead of the IEEE divide sequence
// (v_div_fmas/v_div_fixup chains were serializing the WMMA K-loop).
__device__ __forceinline__ float silu_f(float x) {
    return x * __builtin_amdgcn_rcpf(1.0f + __expf(-x));
}

// ---- K0: spline_weight [i][o][g] -> Wt [(i*17+g)][o]  (278 KB, trivial) -----
__global__ void kan_transpose_w(const float* __restrict__ sw, float* __restrict__ wt) {
    int idx = blockIdx.x * 256 + threadIdx.x;          // 64*17*64 = 69632 total
    int o  = idx & 63;
    int ig = idx >> 6;                                 // i*17 + g
    int i  = ig / GRIDSZ;
    int g  = ig - i * GRIDSZ;
    wt[idx] = sw[(i * OUT_F + o) * GRIDSZ + g];        // coalesced store
}

// ---- K1: basis [B][in][17] == [B][1088], one wave per batch row -------------
__global__ void kan_basis(const float* __restrict__ x, float* __restrict__ basis) {
    __shared__ float xs[8 * IN_F];
    int wave = threadIdx.x >> 5;
    int lane = threadIdx.x & 31;
    int b    = blockIdx.x * 8 + wave;
    // stage input row in LDS
    xs[wave * IN_F + lane]      = x[b * IN_F + lane];
    xs[wave * IN_F + lane + 32] = x[b * IN_F + lane + 32];
    __syncthreads();
    float* brow = basis + (size_t)b * KSP;
    #pragma unroll 2
    for (int c = 0; c < KSP / 32; ++c) {               // 34 coalesced dword sweeps
        int e = c * 32 + lane;
        int i = e / GRIDSZ;
        int g = e - i * GRIDSZ;
        float xv = xs[wave * IN_F + i];
        float gk = GRID_LO + H * (float)g;
        brow[e]  = fmaxf(1.0f - fabsf(xv - gk) * INV_H, 0.0f);
    }
}

// ---- K2: edge_output [B][in][out], one float4 per thread (b128 stores) ------
__global__ void kan_edge(const float* __restrict__ x, const float* __restrict__ bs,
                         const float* __restrict__ wt, float* __restrict__ edge) {
    int idx = blockIdx.x * 256 + threadIdx.x;          // 8192*64*16 float4 elements
    int o4  = idx & 15;
    int i   = (idx >> 4) & 63;
    int b   = idx >> 10;

    float xv = x[b * IN_F + i];                        // broadcast within 16 lanes
    float s  = silu_f(xv);
    // hat basis has exactly <=2 nonzeros: grid cells j, j+1
    float u  = (xv - GRID_LO) * INV_H;
    float fj = floorf(u);
    int   j  = (int)fj;
    float t  = u - fj;
    float w0 = (j >= 0      && j <= GRIDSZ - 1) ? (1.0f - t) : 0.0f;
    float w1 = (j >= -1     && j <= GRIDSZ - 2) ? t          : 0.0f;
    int   g0 = min(max(j, 0), GRIDSZ - 1);
    int   g1 = min(max(j + 1, 0), GRIDSZ - 1);

    const float4* bs4 = (const float4*)bs;
    const float4* wt4 = (const float4*)wt;
    float4 bv = bs4[i * 16 + o4];
    float4 wa = wt4[(i * GRIDSZ + g0) * 16 + o4];
    float4 wb = wt4[(i * GRIDSZ + g1) * 16 + o4];
    float4 e;
    e.x = s * bv.x + w0 * wa.x + w1 * wb.x;
    e.y = s * bv.y + w0 * wa.y + w1 * wb.y;
    e.z = s * bv.z + w0 * wa.z + w1 * wb.z;
    e.w = s * bv.w + w0 * wa.w + w1 * wb.w;
    ((float4*)edge)[idx] = e;
}

// ---- K3: output[b,o] via fp32 WMMA: (silu(X)@BS + basis@Wt)/8 + bias --------
// V_WMMA_F32_16X16X4_F32 layouts (ISA 7.12.2):
//   A (16x4): lane L -> row L%16, VGPR0/1 = K = 2*(L/16), 2*(L/16)+1
//   B (4x16): lane L -> col L%16, VGPR0/1 = same K pair
//   C (16x16): lane L -> col L%16, VGPR v -> row v + 8*(L/16)
__global__ void kan_out_wmma(const float* __restrict__ x, const float* __restrict__ bs,
                             const float* __restrict__ wt, const float* __restrict__ basis,
                             const float* __restrict__ bias, float* __restrict__ out) {
    int lane  = threadIdx.x & 31;
    int wave  = threadIdx.x >> 5;
    int tile  = blockIdx.x * 8 + wave;                 // 512*4 = 2048 tiles of 16x16
    int tm    = tile >> 2;                             // batch tile
    int tn    = tile & 3;                              // out-col tile
    int mrow  = lane & 15;
    int khalf = lane >> 4;                             // 0 or 1
    int kb    = khalf * 2;
    int n     = tn * 16 + (lane & 15);

    v8f c = {};

    // SiLU residual path: K = 64 over inputs, B = base_scale [in][out]
    const float* xrow = x + (tm * 16 + mrow) * IN_F;
    #pragma unroll 4
    for (int k0 = 0; k0 < IN_F; k0 += 4) {
        v2f a;
        a.x = silu_f(xrow[k0 + kb]);
        a.y = silu_f(xrow[k0 + kb + 1]);
        v2f bv;
        bv.x = bs[(k0 + kb)     * OUT_F + n];
        bv.y = bs[(k0 + kb + 1) * OUT_F + n];
        c = __builtin_amdgcn_wmma_f32_16x16x4_f32(false, a, false, bv,
                                                  (short)0, c, false, false);
    }

    // Spline path: K = 1088 over basis, B = Wt [(i*17+g)][out]
    for (int k0 = 0; k0 < KSP; k0 += 4) {
        const float* arow = basis + (size_t)(tm * 16 + mrow) * KSP;
        v2f a = *(const v2f*)(arow + k0 + kb);         // 8B-aligned (row stride 4352B)
        v2f bv;
        bv.x = wt[(k0 + kb)     * OUT_F + n];
        bv.y = wt[(k0 + kb + 1) * OUT_F + n];
        c = __builtin_amdgcn_wmma_f32_16x16x4_f32(false, a, false, bv,
                                                  (short)0, c, false, false);
    }

    float bb = bias[n];
    #pragma unroll
    for (int v = 0; v < 8; ++v) {
        int row = tm * 16 + v + 8 * khalf;
        out[row * OUT_F + n] = c[v] * INV_SQRT_IN + bb;
    }
}

// ---- host-side launcher -----------------------------------------------------
extern "C" void kernel_launch(void* const* d_in, const int* in_sizes, int n_in,
                              void* d_out, int out_size, void* d_ws, size_t ws_size,
                              hipStream_t stream) {
    (void)in_sizes; (void)n_in; (void)out_size; (void)ws_size;
    const float* x    = (const float*)d_in[0];   // [8192,64]
    const float* bsc  = (const float*)d_in[1];   // [64,64]
    const float* sw   = (const float*)d_in[2];   // [64,64,17]
    const float* bias = (const float*)d_in[3];   // [64]

    float* out   = (float*)d_out;                        // [8192,64]
    float* edge  = out  + (size_t)BATCH * OUT_F;         // [8192,64,64]
    float* basis = edge + (size_t)BATCH * IN_F * OUT_F;  // [8192,64,17] == [8192,1088]
    float* wt    = (float*)d_ws;                         // [1088,64] transposed weights

    kan_transpose_w<<<(IN_F * OUT_F * GRIDSZ) / 256, 256, 0, stream>>>(sw, wt);
    kan_basis      <<<BATCH / 8,                    256, 0, stream>>>(x, basis);
    kan_edge       <<<(BATCH * IN_F * 16) / 256,    256, 0, stream>>>(x, bsc, wt, edge);
    kan_out_wmma   <<<(BATCH / 16) * 4 / 8,         256, 0, stream>>>(x, bsc, wt, basis, bias, out);
}